// ScaleLasryLions_33131377722051
// MI455X (gfx1250) — compile-verified
//
#include <hip/hip_runtime.h>
#include <stdint.h>

// Problem constants (from reference setup_inputs / _scale_params)
static constexpr int B_ = 8, H_ = 192, W_ = 192, S_ = 4, C_ = 32;
static constexpr int TP  = 96;     // output line positions per block
static constexpr int RPT = 12;     // outputs per thread (8 waves * 12 = 96)
static constexpr int LMAX = 37;    // 2*18+1
static constexpr int LDS_PIX = TP + LMAX - 1 + 4;  // 136 pixel slots

__device__ __forceinline__ int imind(int a, int b) { return a < b ? a : b; }
__device__ __forceinline__ int imaxd(int a, int b) { return a > b ? a : b; }

__device__ __forceinline__ void async_load16(unsigned lds_off, unsigned long long gaddr) {
  // GLOBAL_LOAD_ASYNC_TO_LDS_B128: per-lane 16B global -> LDS, tracked by ASYNCcnt
  asm volatile("global_load_async_to_lds_b128 %0, %1, off"
               :: "v"(lds_off), "v"(gaddr) : "memory");
}

__device__ __forceinline__ void wait_async0() {
#if __has_builtin(__builtin_amdgcn_s_wait_asynccnt)
  __builtin_amdgcn_s_wait_asynccnt(0);
#else
  asm volatile("s_wait_asynccnt 0" ::: "memory");
#endif
}

// Fully-unrolled tap steps j = J .. LT-1 with a circular register window.
// buf[(j+i) % RPT] holds lds[plb + j + i]; indices fold to constants at each
// instantiation, so the rotation is pure register renaming (no moves, no spill).
template <int J, int LT, bool DILATE>
struct Steps {
  __device__ static __forceinline__ void run(const float* __restrict__ ldsb,
                                             const float* __restrict__ fbs,
                                             float coefc, float cc,
                                             float (&buf)[RPT], float (&acc)[RPT]) {
    buf[(J - 1) % RPT] = ldsb[(J + RPT - 1) * C_];   // one new value enters window
    const float t = (fbs[J] * coefc) * cc;
#pragma unroll
    for (int ii = 0; ii < RPT; ++ii) {
      const float v = buf[(J + ii) % RPT];
      acc[ii] = DILATE ? fmaxf(acc[ii], v + t) : fminf(acc[ii], v - t);
    }
    Steps<J + 1, LT, DILATE>::run(ldsb, fbs, coefc, cc, buf, acc);
  }
};
template <int LT, bool DILATE>
struct Steps<LT, LT, DILATE> {
  __device__ static __forceinline__ void run(const float*, const float*, float, float,
                                             float (&)[RPT], float (&)[RPT]) {}
};

// One separable morphological pass.
//  AXIS_H : pass runs along H (line coordinate is a row index), else along W
//  DILATE : max-plus (v + tap); else min-minus (v - tap)
//  SRC_X  : read from the 5-D input x with edge clamping (implies AXIS_H)
//  DSTMODE: 0 = dense scratch buffer, 1 = raw xc into d_out slot (axis W only),
//           2 = alpha blend with xc already in d_out (axis W only)
//  USE_C  : taps are kc = (fb*coef)*c instead of k = fb*coef
//  LT     : tap count (compile time), w = (LT-1)/2
template <bool AXIS_H, bool DILATE, bool SRC_X, int DSTMODE, bool USE_C, int LT>
__global__ __launch_bounds__(256)
void morph_pass(const float* __restrict__ src, float* __restrict__ dst,
                const float* __restrict__ coef, const float* __restrict__ cvec,
                const float* __restrict__ alpha,
                int scaleIdx, float s2f,
                int srcSH, int srcSW, int lineLen, int crossLen) {
  constexpr int w = (LT - 1) / 2;
  __shared__ float lds[LDS_PIX * C_];
  __shared__ float fbs[LT];

  const int tid = threadIdx.x;
  const int c   = tid & 31;   // channel == lane
  const int r   = tid >> 5;   // wave index 0..7
  const int p0  = blockIdx.x * TP;
  const int cross = blockIdx.y;
  const int b     = blockIdx.z;

  const int tileLen = imind(TP, lineLen - p0);
  const int NP      = tileLen + LT - 1;       // pixels to stage
  const int NPc     = (NP + 3) & ~3;          // round to 4-pixel chunks

  // ---- per-block taps base: fb[j] = -(j-w)^2 / s^2 (exact reference rounding) ----
  if (tid < LT) {
    const int d = tid - w;
    fbs[tid] = -((float)(d * d)) / s2f;
  }

  // ---- stage NP pixels (128B each) into LDS via async copy engine ----
  const int lanePix  = c >> 3;          // which of 4 pixels in a 512B wave chunk
  const int laneByte = (c & 7) * 16;    // 16B slice within the 128B pixel
  const unsigned ldsBase = (unsigned)(uintptr_t)(const void*)lds;
  const unsigned long long gBase = (unsigned long long)(uintptr_t)src;

  for (int ch4 = r * 4; ch4 < NPc; ch4 += 32) {   // wave r takes chunks r, r+8, ...
    const int ql = ch4 + lanePix;                  // LDS pixel slot
    const int q  = p0 + imind(ql, NP - 1);         // clamped line coordinate
    unsigned pixOff;                               // float offset of pixel start
    if (SRC_X) {
      const int hsrc = imaxd(0, imind(q - 2 * w, H_ - 1));
      const int wsrc = imaxd(0, imind(cross - 2 * w, W_ - 1));
      pixOff = (unsigned)((((b * H_ + hsrc) * W_ + wsrc) * S_ + scaleIdx) * C_);
    } else if (AXIS_H) {
      pixOff = (unsigned)(((b * srcSH + q) * srcSW + cross) * C_);
    } else {
      pixOff = (unsigned)(((b * srcSH + cross) * srcSW + q) * C_);
    }
    const unsigned long long ga = gBase + (unsigned long long)(pixOff * 4u + (unsigned)laneByte);
    const unsigned la = ldsBase + (unsigned)(ql * (C_ * 4) + laneByte);
    async_load16(la, ga);
  }
  wait_async0();
  __syncthreads();

  const float coefc = coef[c];
  const float cc    = USE_C ? cvec[c] : 1.0f;   // *1.0f is exact => matches (fb*coef)*c

  // ---- sliding-window max-plus / min-minus: RPT+LT-1 LDS reads total ----
  const int plb = r * RPT;
  const float* __restrict__ ldsb = &lds[plb * C_ + c];
  float buf[RPT], acc[RPT];
#pragma unroll
  for (int ii = 0; ii < RPT; ++ii) buf[ii] = ldsb[ii * C_];
  {
    const float t = (fbs[0] * coefc) * cc;
#pragma unroll
    for (int ii = 0; ii < RPT; ++ii) acc[ii] = DILATE ? (buf[ii] + t) : (buf[ii] - t);
  }
  Steps<1, LT, DILATE>::run(ldsb, fbs, coefc, cc, buf, acc);

  // ---- store ----
#pragma unroll
  for (int ii = 0; ii < RPT; ++ii) {
    const int pl = plb + ii;
    if (pl >= tileLen) break;
    const int p = p0 + pl;
    if (DSTMODE == 0) {
      const unsigned off = AXIS_H
          ? (unsigned)(((b * lineLen + p) * crossLen + cross) * C_)
          : (unsigned)(((b * crossLen + cross) * lineLen + p) * C_);
      dst[off + c] = acc[ii];
    } else {
      const unsigned off = (unsigned)((((b * H_ + cross) * W_ + p) * S_ + scaleIdx) * C_ + c);
      if (DSTMODE == 1) {
        dst[off] = acc[ii];                       // raw xc parked in d_out
      } else {
        const float a = alpha[c];
        dst[off] = a * dst[off] + (1.0f - a) * acc[ii];   // alpha*xc + (1-alpha)*xo
      }
    }
  }
}

template <int LT>
static void launch_scale(const float* x, float* out, float* bufA, float* bufB,
                         const float* coef, const float* cvec, const float* alpha,
                         int i, float s2, hipStream_t stream) {
  constexpr int w   = (LT - 1) / 2;
  constexpr int HP  = H_ + 2 * w;   // rows after first H pass
  constexpr int WP1 = W_ + 4 * w;   // padded column count
  constexpr int WP2 = W_ + 2 * w;   // cols after first W pass
  const dim3 blk(256);
  auto grd = [](int lineLen, int crossLen) {
    return dim3((unsigned)((lineLen + TP - 1) / TP), (unsigned)crossLen, (unsigned)B_);
  };

  // ---- closing chain (xc): dilate k (H,W) then erode kc (H,W) ----
  morph_pass<true,  true,  true,  0, false, LT><<<grd(HP,  WP1), blk, 0, stream>>>(
      x,    bufA, coef, cvec, alpha, i, s2, 0,   0,   HP,  WP1);
  morph_pass<false, true,  false, 0, false, LT><<<grd(WP2, HP ), blk, 0, stream>>>(
      bufA, bufB, coef, cvec, alpha, i, s2, HP,  WP1, WP2, HP);
  morph_pass<true,  false, false, 0, true,  LT><<<grd(H_,  WP2), blk, 0, stream>>>(
      bufB, bufA, coef, cvec, alpha, i, s2, HP,  WP2, H_,  WP2);
  morph_pass<false, false, false, 1, true,  LT><<<grd(W_,  H_ ), blk, 0, stream>>>(
      bufA, out,  coef, cvec, alpha, i, s2, H_,  WP2, W_,  H_);

  // ---- opening chain (xo): erode k (H,W) then dilate kc (H,W), blend ----
  morph_pass<true,  false, true,  0, false, LT><<<grd(HP,  WP1), blk, 0, stream>>>(
      x,    bufA, coef, cvec, alpha, i, s2, 0,   0,   HP,  WP1);
  morph_pass<false, false, false, 0, false, LT><<<grd(WP2, HP ), blk, 0, stream>>>(
      bufA, bufB, coef, cvec, alpha, i, s2, HP,  WP1, WP2, HP);
  morph_pass<true,  true,  false, 0, true,  LT><<<grd(H_,  WP2), blk, 0, stream>>>(
      bufB, bufA, coef, cvec, alpha, i, s2, HP,  WP2, H_,  WP2);
  morph_pass<false, true,  false, 2, true,  LT><<<grd(W_,  H_ ), blk, 0, stream>>>(
      bufA, out,  coef, cvec, alpha, i, s2, H_,  WP2, W_,  H_);
}

extern "C" void kernel_launch(void* const* d_in, const int* in_sizes, int n_in,
                              void* d_out, int out_size, void* d_ws, size_t ws_size,
                              hipStream_t stream) {
  (void)in_sizes; (void)n_in; (void)out_size; (void)ws_size;
  const float* x     = (const float*)d_in[0];
  const float* coef  = (const float*)d_in[1];
  const float* cvec  = (const float*)d_in[2];
  const float* alpha = (const float*)d_in[3];
  float* out = (float*)d_out;

  // scratch ping-pong buffers (sized for the largest scale, w=18)
  float* bufA = (float*)d_ws;                                   // (B, H+2w, W+4w, C)
  const size_t bufAElems = (size_t)B_ * (H_ + 36) * (W_ + 72) * C_;
  float* bufB = bufA + bufAElems;                               // (B, H+2w, W+2w, C)

  // scales = (2^(i+1) - 1)*0.25 -> s^2 in exact f32
  launch_scale<3 >(x, out, bufA, bufB, coef, cvec, alpha, 0, 0.0625f,  stream);
  launch_scale<7 >(x, out, bufA, bufB, coef, cvec, alpha, 1, 0.5625f,  stream);
  launch_scale<17>(x, out, bufA, bufB, coef, cvec, alpha, 2, 3.0625f,  stream);
  launch_scale<37>(x, out, bufA, bufB, coef, cvec, alpha, 3, 14.0625f, stream);
}